// WaveCell_69011534512145
// MI455X (gfx1250) — compile-verified
//
#include <hip/hip_runtime.h>

// Fused 2D staggered-grid elastic wave step (velocity-stress + damping).
// Memory-bound: ~0.74 GB min traffic -> ~32us @ 23.3 TB/s. Strategy:
//  * stage stress tiles (halo 2) into LDS via CDNA5 async global->LDS DMA
//  * compute vx_new/vz_new on a halo-1 extended tile, keep them in LDS
//  * fused stress update reads velocities + old stresses from LDS only
// Output layout: [5, B, NZ, NX] = stack(vx, vz, txx, tzz, txz).

#define DT_C 1.0e-3f
#define H_C  10.0f

constexpr int TX = 64;           // tile width (x, contiguous)
constexpr int TZ = 16;           // tile height (z)
constexpr int SW = TX + 4;       // stress tile width  (halo 2)  = 68
constexpr int SH = TZ + 4;       // stress tile height (halo 2)  = 20
constexpr int VW = TX + 2;       // velocity tile width  (halo 1) = 66
constexpr int VH = TZ + 2;       // velocity tile height (halo 1) = 18

__device__ __forceinline__ int wrapi(int v, int n) {
    if (v < 0)  return v + n;
    if (v >= n) return v - n;
    return v;
}

// Per-lane async copy of 4 bytes: from the global address in gaddr to the LDS address in lds_off. Tracked by ASYNCcnt.
__device__ __forceinline__ void async_g2l_b32(unsigned int lds_off,
                                              unsigned long long gaddr) {
    asm volatile("global_load_async_to_lds_b32 %0, %1, off"
                 :: "v"(lds_off), "v"(gaddr) : "memory");
}

__global__ __launch_bounds__(256) void WaveCell_69011534512145_kernel(
    const float* __restrict__ vp,  const float* __restrict__ vs,
    const float* __restrict__ rho,
    const float* __restrict__ vx,  const float* __restrict__ vz,
    const float* __restrict__ txx, const float* __restrict__ tzz,
    const float* __restrict__ txz,
    const float* __restrict__ d,
    float* __restrict__ out, int NZ, int NX, int B)
{
    __shared__ float s_txx[SH * SW];
    __shared__ float s_tzz[SH * SW];
    __shared__ float s_txz[SH * SW];
    __shared__ float s_vx [VH * VW];
    __shared__ float s_vz [VH * VW];

    const int x0 = blockIdx.x * TX;
    const int z0 = blockIdx.y * TZ;
    const int b  = blockIdx.z;
    const size_t plane = (size_t)NZ * (size_t)NX;
    const size_t fbase = (size_t)b * plane;
    const int tid = threadIdx.y * blockDim.x + threadIdx.x;   // 0..255

    const float* gtxx = txx + fbase;
    const float* gtzz = tzz + fbase;
    const float* gtxz = txz + fbase;
    const float* gvx  = vx  + fbase;
    const float* gvz  = vz  + fbase;

    // ---- Stage 0: async DMA stress tiles (halo 2) into LDS ------------------
    for (int i = tid; i < SH * SW; i += 256) {
        const int ez = i / SW - 2;
        const int ex = i % SW - 2;
        const int gz = wrapi(z0 + ez, NZ);
        const int gx = wrapi(x0 + ex, NX);
        const size_t goff = (size_t)gz * NX + gx;
        async_g2l_b32((unsigned int)(unsigned long long)(const void*)&s_txx[i],
                      (unsigned long long)(const void*)(gtxx + goff));
        async_g2l_b32((unsigned int)(unsigned long long)(const void*)&s_tzz[i],
                      (unsigned long long)(const void*)(gtzz + goff));
        async_g2l_b32((unsigned int)(unsigned long long)(const void*)&s_txz[i],
                      (unsigned long long)(const void*)(gtxz + goff));
    }
    asm volatile("s_wait_asynccnt 0" ::: "memory");   // my wave's DMA done
    __syncthreads();                                  // everyone's DMA visible

    // ---- Stage 1: velocity update on halo-1 extended tile -------------------
    float* ovx = out + 0 * (size_t)B * plane + fbase;
    float* ovz = out + 1 * (size_t)B * plane + fbase;
    for (int i = tid; i < VH * VW; i += 256) {
        const int ez = i / VW - 1;
        const int ex = i % VW - 1;
        const int gz = wrapi(z0 + ez, NZ);
        const int gx = wrapi(x0 + ex, NX);
        const size_t goff = (size_t)gz * NX + gx;

        const float dd  = d[goff];
        const float rh  = rho[goff];
        const float cc  = 0.5f * DT_C * dd;
        const float inv = 1.0f / (1.0f + cc);
        const float aa  = (1.0f - cc) * inv;
        const float br  = (DT_C * inv) / (H_C * rh);   // b / (H * rho)

        const int si = (ez + 2) * SW + (ex + 2);
        const float txxc = s_txx[si];
        const float tzzc = s_tzz[si];
        const float txzc = s_txz[si];
        // vx_new = a*vx + b*(dminus_x(txx) + dminus_z(txz))/rho
        const float vxn = aa * gvx[goff]
            + br * ((txxc - s_txx[si - 1]) + (txzc - s_txz[si - SW]));
        // vz_new = a*vz + b*(dplus_x(txz) + dplus_z(tzz))/rho
        const float vzn = aa * gvz[goff]
            + br * ((s_txz[si + 1] - txzc) + (s_tzz[si + SW] - tzzc));

        s_vx[i] = vxn;
        s_vz[i] = vzn;
        if (ez >= 0 && ez < TZ && ex >= 0 && ex < TX &&
            (z0 + ez) < NZ && (x0 + ex) < NX) {
            ovx[goff] = vxn;
            ovz[goff] = vzn;
        }
    }
    __syncthreads();

    // ---- Stage 2: stress update from LDS velocities + LDS old stresses ------
    const int tx = threadIdx.x;
    const float rH = 1.0f / H_C;
    float* otxx = out + 2 * (size_t)B * plane + fbase;
    float* otzz = out + 3 * (size_t)B * plane + fbase;
    float* otxz = out + 4 * (size_t)B * plane + fbase;
    #pragma unroll
    for (int k = 0; k < TZ / 4; ++k) {
        const int tz = (int)threadIdx.y + 4 * k;
        const int gz = z0 + tz;
        const int gx = x0 + tx;
        if (gz >= NZ || gx >= NX) continue;
        const size_t goff = (size_t)gz * NX + gx;

        const int vi = (tz + 1) * VW + (tx + 1);
        const float vxc = s_vx[vi];
        const float vzc = s_vz[vi];
        const float vx_x = (s_vx[vi + 1]  - vxc) * rH;            // dplus_x(vx)
        const float vz_z = (vzc - s_vz[vi - VW]) * rH;            // dminus_z(vz)
        const float sxz  = ((s_vx[vi + VW] - vxc) +               // dplus_z(vx)
                            (vzc - s_vz[vi - 1])) * rH;           // + dminus_x(vz)

        const float dd  = d[goff];
        const float rh  = rho[goff];
        const float vpv = vp[goff];
        const float vsv = vs[goff];
        const float mu  = rh * vsv * vsv;
        const float lpm = rh * vpv * vpv;        // lam + 2*mu
        const float lam = lpm - 2.0f * mu;

        const float cc  = 0.5f * DT_C * dd;
        const float inv = 1.0f / (1.0f + cc);
        const float aa  = (1.0f - cc) * inv;
        const float bb  = DT_C * inv;

        const int si = (tz + 2) * SW + (tx + 2);
        otxx[goff] = aa * s_txx[si] + bb * (lpm * vx_x + lam * vz_z);
        otzz[goff] = aa * s_tzz[si] + bb * (lpm * vz_z + lam * vx_x);
        otxz[goff] = aa * s_txz[si] + bb * mu * sxz;
    }
}

extern "C" void kernel_launch(void* const* d_in, const int* in_sizes, int n_in,
                              void* d_out, int out_size, void* d_ws, size_t ws_size,
                              hipStream_t stream) {
    (void)n_in; (void)d_ws; (void)ws_size; (void)out_size;
    const float* vp  = (const float*)d_in[0];
    const float* vs  = (const float*)d_in[1];
    const float* rho = (const float*)d_in[2];
    const float* vx  = (const float*)d_in[3];
    const float* vz  = (const float*)d_in[4];
    const float* txx = (const float*)d_in[5];
    const float* tzz = (const float*)d_in[6];
    const float* txz = (const float*)d_in[7];
    const float* d   = (const float*)d_in[8];
    float* out = (float*)d_out;

    const int plane = in_sizes[0];          // NZ * NX
    int N = 1;
    while ((long long)N * (long long)N < (long long)plane) ++N;  // square grid
    const int NZ = N, NX = N;
    const int B = in_sizes[3] / plane;

    dim3 block(TX, 4, 1);                    // 256 threads = 8 wave32
    dim3 grid((NX + TX - 1) / TX, (NZ + TZ - 1) / TZ, B);
    WaveCell_69011534512145_kernel<<<grid, block, 0, stream>>>(
        vp, vs, rho, vx, vz, txx, tzz, txz, d, out, NZ, NX, B);
}